// upperExtremityModel_83966610637001
// MI455X (gfx1250) — compile-verified
//
#include <hip/hip_runtime.h>
#include <stdint.h>

typedef unsigned int u32x4 __attribute__((ext_vector_type(4)));
typedef int          i32x8 __attribute__((ext_vector_type(8)));
typedef int          i32x4 __attribute__((ext_vector_type(4)));

namespace {
constexpr int kBatch   = 1024;
constexpr int kT       = 2048;
constexpr int kJ       = 8;
constexpr int kCT      = 32;            // time steps staged per TDM chunk
constexpr int kChunks  = kT / kCT;      // 64
constexpr int kBperBlk = 4;             // batches per single-wave block (4*8 joints = 32 lanes)

// Bilinear model constants, pre-folded:
//   omega' = A*omega + C*theta + D
//   A = 1 - DT*B_J/I_J = -0.25
//   C = -(DT/I_J)*(M^2*(Km0+Km1) + K_J)
//   D =  (DT/I_J)*M*L1*(Km1*a1 - Km0*a0)
constexpr float kDT = 1.0f / 60.0f;
constexpr float kA  = -0.25f;
constexpr float kCS = -0.0104166670f;   // -(DT/I_J)*M*M
constexpr float kC0 = -20.8333340f;     // -(DT/I_J)*K_J
constexpr float kDC = 0.00125f;         //  (DT/I_J)*M*L1
}  // namespace

// Issue one 2D TDM transfer: tile = [4 batch rows] x [kCT*16 contiguous floats],
// row stride = kT*16 floats. LDS padding: +8 dwords after every 256 dwords so the
// per-batch LDS stride is 528 dwords (== 16 mod 64 -> bank-conflict-free b64 reads).
__device__ __forceinline__ void tdm_load_chunk(const float* gsrc, unsigned lds_byte_addr) {
  unsigned long long ga = (unsigned long long)(uintptr_t)gsrc;
  u32x4 g0;
  g0.x = 1u;                                                   // count=1 (valid user D#)
  g0.y = lds_byte_addr;                                        // lds_addr [63:32]
  g0.z = (unsigned)ga;                                         // global_addr[31:0]
  g0.w = (unsigned)((ga >> 32) & 0x01FFFFFFu) | (2u << 30);    // global_addr[56:32] | type=2
  i32x8 g1;
  g1[0] = (int)((2u << 16)        // data_size = 4 bytes
              | (1u << 20)        // pad_enable
              | (7u << 22)        // pad_interval = 256 dwords
              | (7u << 25));      // pad_amount   = 8 dwords
  g1[1] = (int)0x80000000u;                          // tensor_dim0 = 32768 (bits[63:48])
  g1[2] = (int)((unsigned)kBperBlk << 16);           // tensor_dim1 = 4    (bits[111:96..80])
  g1[3] = (int)((unsigned)(kCT * 16) << 16);         // tile_dim0 = 512    (bits[127:112])
  g1[4] = kBperBlk;                                  // tile_dim1 = 4, tile_dim2 = 0
  g1[5] = kT * 16;                                   // tensor_dim0_stride = 32768 elements
  g1[6] = 0;                                         // stride hi / dim1_stride lo
  g1[7] = 0;
  i32x4 z4 = {0, 0, 0, 0};
#if __clang_major__ >= 23
  i32x8 z8 = {0, 0, 0, 0, 0, 0, 0, 0};
  __builtin_amdgcn_tensor_load_to_lds(g0, g1, z4, z4, z8, 0);
#else
  __builtin_amdgcn_tensor_load_to_lds(g0, g1, z4, z4, 0);
#endif
}

__global__ __launch_bounds__(32) void muscle_scan_kernel(
    const float* __restrict__ x, const float* __restrict__ st0,
    float* __restrict__ out, float* __restrict__ stf) {
  // [buf][batch_local][halfchunk][256 data + 8 pad dwords]
  __shared__ __align__(16) float sbuf[2][kBperBlk][2][264];

  const int lane = threadIdx.x;
  const int bl   = lane >> 3;   // batch within block: 0..3
  const int j    = lane & 7;    // joint: 0..7
  const int bk   = blockIdx.x;
  const int b    = bk * kBperBlk + bl;

  float theta = st0[b * 16 + 2 * j];
  float omega = st0[b * 16 + 2 * j + 1];

  const float* gbase = x + (size_t)bk * kBperBlk * ((size_t)kT * 16);
  float*       op    = out + (size_t)b * kT * kJ + j;

  // Prologue: stage chunk 0.
  tdm_load_chunk(gbase, (unsigned)(uintptr_t)&sbuf[0][0][0][0]);

  for (int c = 0; c < kChunks; ++c) {
    __builtin_amdgcn_s_wait_tensorcnt(0);      // chunk c resident in LDS
    asm volatile("" ::: "memory");             // fence compiler reordering of ds loads
    if (c + 1 < kChunks) {                     // overlap next transfer with compute
      tdm_load_chunk(gbase + (size_t)(c + 1) * (kCT * 16),
                     (unsigned)(uintptr_t)&sbuf[(c + 1) & 1][0][0][0]);
    }
    const float* buf = &sbuf[c & 1][bl][0][0];
#pragma unroll 8
    for (int t = 0; t < kCT; ++t) {
      const float2 a =
          *(const float2*)(buf + (t >> 4) * 264 + ((t & 15) << 4) + (j << 1));
      // Off-critical-path input coefficients:
      float Km0  = fmaf(2000.0f, a.x, 100.0f);
      float Km1  = fmaf(2000.0f, a.y, 100.0f);
      float ssum = Km0 + Km1;
      float diff = fmaf(Km1, a.y, -(Km0 * a.x));   // Km1*a1 - Km0*a0
      float C    = fmaf(kCS, ssum, kC0);
      float D    = kDC * diff;
      // Critical path: 2 dependent FMAs, then theta (parallel-ish).
      omega = fmaf(kA, omega, fmaf(C, theta, D));
      theta = fmaf(kDT, omega, theta);
      __builtin_nontemporal_store(theta, op + (size_t)(c * kCT + t) * kJ);
    }
  }

  // Final states: [b][2j]=theta, [b][2j+1]=omega
  stf[b * 16 + 2 * j]     = theta;
  stf[b * 16 + 2 * j + 1] = omega;
}

extern "C" void kernel_launch(void* const* d_in, const int* in_sizes, int n_in,
                              void* d_out, int out_size, void* d_ws, size_t ws_size,
                              hipStream_t stream) {
  (void)in_sizes; (void)n_in; (void)out_size; (void)d_ws; (void)ws_size;
  const float* x  = (const float*)d_in[0];
  const float* s0 = (const float*)d_in[1];
  float* out = (float*)d_out;
  float* stf = out + (size_t)kBatch * kT * kJ;
  dim3 grid(kBatch / kBperBlk);  // 256 single-wave workgroups
  dim3 block(32);
  hipLaunchKernelGGL(muscle_scan_kernel, grid, block, 0, stream, x, s0, out, stf);
}